// GraphConv_76630806495330
// MI455X (gfx1250) — compile-verified
//
#include <hip/hip_runtime.h>

#define N_NODES 100000
#define N_EDGES 3200000
#define D_IN    512
#define D_OUT   256

typedef __attribute__((ext_vector_type(2))) float v2f;
typedef __attribute__((ext_vector_type(8))) float v8f;

// ---------------------------------------------------------------------------
// Kernel 1: xw = x @ W using V_WMMA_F32_16X16X4_F32.
// One wave per 16-row M tile; all 16 N tiles (256 cols) accumulated in
// registers (16 x v8f) so x streams from HBM exactly once. W (512KB) is
// L2-resident and re-read per wave.
// A layout (16x4 f32): lanes 0-15 -> K=0,1 ; lanes 16-31 -> K=2,3 (v2f).
// B layout (4x16 f32): VGPR0 = row K (=khalf), VGPR1 = row K+1, N striped
// across lane%16.
// C/D layout: VGPR r -> M = r + (lane>=16 ? 8 : 0), N = lane%16.
// ---------------------------------------------------------------------------
__global__ __launch_bounds__(128)
void gemm_xw_wmma(const float* __restrict__ x,
                  const float* __restrict__ w,
                  float* __restrict__ xw) {
  const int wave = blockIdx.x * (blockDim.x >> 5) + (threadIdx.x >> 5);
  const int lane = threadIdx.x & 31;
  const int m0 = wave * 16;
  if (m0 >= N_NODES) return;

  const int l16   = lane & 15;
  const int khalf = (lane >> 4) * 2;        // 0 for lanes 0-15, 2 for 16-31

  v8f acc[16];
#pragma unroll
  for (int t = 0; t < 16; ++t) acc[t] = (v8f){0,0,0,0,0,0,0,0};

  const float* __restrict__ xrow = x + (size_t)(m0 + l16) * D_IN;

  for (int k = 0; k < D_IN; k += 4) {
    const int kk = k + khalf;
    // contiguous, 8B-aligned pair from row-major x -> global_load_b64
    const v2f a = *(const v2f*)(xrow + kk);

#pragma unroll
    for (int t = 0; t < 16; ++t) {
      v2f b;
      b.x = w[(size_t)kk       * D_OUT + t * 16 + l16];
      b.y = w[(size_t)(kk + 1) * D_OUT + t * 16 + l16];
      acc[t] = __builtin_amdgcn_wmma_f32_16x16x4_f32(
          /*neg_a=*/false, a, /*neg_b=*/false, b,
          /*c_mod=*/(short)0, acc[t], /*reuse_a=*/false, /*reuse_b=*/false);
    }
  }

  const int mbase = m0 + ((lane >> 4) * 8);
#pragma unroll
  for (int t = 0; t < 16; ++t) {
#pragma unroll
    for (int r = 0; r < 8; ++r) {
      xw[(size_t)(mbase + r) * D_OUT + t * 16 + l16] = acc[t][r];
    }
  }
}

// ---------------------------------------------------------------------------
// Native fp32 atomic add, no return (STOREcnt-tracked). Guarantees
// global_atomic_add_f32 instead of a possible CAS-loop legalization.
// ---------------------------------------------------------------------------
__device__ __forceinline__ void atomic_add_f32(float* p, float v) {
  asm volatile("global_atomic_add_f32 %0, %1, off"
               :
               : "v"(p), "v"(v)
               : "memory");
}

// ---------------------------------------------------------------------------
// Kernel 2: edge scatter. One wave per edge: gather xw[col] (256 floats as
// 2 x float4 per lane, coalesced b128), scale by edge_val, atomic-add into
// out[row]. xw (102MB) and out (102MB) both fit in the 192MB L2, so the
// random gather/atomic traffic should mostly stay on-chip.
// Edge metadata is wave-uniform: force the index through readfirstlane so
// row/col/val become scalar (SGPR) loads.
// ---------------------------------------------------------------------------
__global__ __launch_bounds__(256)
void scatter_edges(const float* __restrict__ xw,
                   const int*   __restrict__ erow,
                   const int*   __restrict__ ecol,
                   const float* __restrict__ eval_,
                   float* __restrict__ out) {
  const int e = __builtin_amdgcn_readfirstlane(
      blockIdx.x * (blockDim.x >> 5) + (threadIdx.x >> 5));
  const int lane = threadIdx.x & 31;
  if (e >= N_EDGES) return;

  const int   row = erow[e];
  const int   col = ecol[e];
  const float v   = eval_[e];

  const float4* __restrict__ src = (const float4*)(xw + (size_t)col * D_OUT);
  float* __restrict__ dst = out + (size_t)row * D_OUT;

#pragma unroll
  for (int h = 0; h < 2; ++h) {
    const int q = lane + h * 32;          // float4 index 0..63
    float4 d = src[q];
    float* p = dst + q * 4;
    atomic_add_f32(p + 0, v * d.x);
    atomic_add_f32(p + 1, v * d.y);
    atomic_add_f32(p + 2, v * d.z);
    atomic_add_f32(p + 3, v * d.w);
  }
}

// ---------------------------------------------------------------------------
// Kernel 3: out = relu(out + bias), in place, float4-vectorized.
// ---------------------------------------------------------------------------
__global__ __launch_bounds__(256)
void bias_relu(float* __restrict__ out, const float* __restrict__ bias) {
  const size_t i = (size_t)blockIdx.x * blockDim.x + threadIdx.x;  // float4 idx
  const size_t total4 = (size_t)N_NODES * D_OUT / 4;
  if (i >= total4) return;

  float4 vv = ((const float4*)out)[i];
  const float4 b = ((const float4*)bias)[i & (D_OUT / 4 - 1)];
  vv.x = fmaxf(vv.x + b.x, 0.0f);
  vv.y = fmaxf(vv.y + b.y, 0.0f);
  vv.z = fmaxf(vv.z + b.z, 0.0f);
  vv.w = fmaxf(vv.w + b.w, 0.0f);
  ((float4*)out)[i] = vv;
}

// ---------------------------------------------------------------------------
extern "C" void kernel_launch(void* const* d_in, const int* in_sizes, int n_in,
                              void* d_out, int out_size, void* d_ws, size_t ws_size,
                              hipStream_t stream) {
  const float* x     = (const float*)d_in[0];
  const int*   erow  = (const int*)  d_in[1];
  const int*   ecol  = (const int*)  d_in[2];
  const float* eval_ = (const float*)d_in[3];
  const float* w     = (const float*)d_in[4];
  const float* bias  = (const float*)d_in[5];
  float*       out   = (float*)d_out;
  float*       xw    = (float*)d_ws;   // 100000*256*4 = 102.4 MB scratch

  // Zero the aggregation target (capture-legal, stream-ordered).
  hipMemsetAsync(out, 0, (size_t)N_NODES * D_OUT * sizeof(float), stream);

  // GEMM: 6250 M-tiles, 4 waves (128 threads) per block.
  const int gemm_waves  = N_NODES / 16;                 // 6250
  const int gemm_blocks = (gemm_waves + 3) / 4;         // 1563
  gemm_xw_wmma<<<gemm_blocks, 128, 0, stream>>>(x, w, xw);

  // Scatter: one wave per edge, 8 waves (256 threads) per block.
  const int scat_blocks = (N_EDGES + 7) / 8;            // 400000
  scatter_edges<<<scat_blocks, 256, 0, stream>>>(xw, erow, ecol, eval_, out);

  // Bias + ReLU in place.
  const size_t total4 = (size_t)N_NODES * D_OUT / 4;
  bias_relu<<<(int)((total4 + 255) / 256), 256, 0, stream>>>(out, bias);
}